// SkinnyQuadTModel_77962246357369
// MI455X (gfx1250) — compile-verified
//
#include <hip/hip_runtime.h>

// ---------------------------------------------------------------------------
// out[r] = -(x_r . v)^2 + x_r . w + b          (B=16384 rows, D=4096 cols)
//
// HBM-bound: 256 MB fp32 in vs ~268 MFLOP -> roofline ~11us @ 23.3 TB/s.
//   * A (16x4 fp32) = [v_chunk ; w_chunk ; zeros...]   (params)
//   * B (4x16 fp32) = x^T tile (16 input rows as columns)
//   * D = AxB + C  => acc VGPR0 lanes0-15 = v.x_n, VGPR1 lanes0-15 = w.x_n
//   * Coalesced global b128 -> LDS staging, double buffered; chunk loop is
//     unrolled by 2 so the buffer index is compile-time constant and every
//     LDS access is a direct GEP on the __shared__ arrays (keeps addrspace(3)
//     => real ds_* ops, not flat_*).
//   * pitch 68 dwords => conflict-free ds_load_b64 fragment gathers;
//     wave-private LDS => no barriers (DS ops are in-order per wave).
//   * A-fragment gather is UNCONDITIONAL: rows 2..15 read a zero-filled LDS
//     region via a per-lane base pointer (no EXEC masking, no selects in loop).
//   * K split into 4 segments (4096 waves); deterministic partials in d_ws.
// ---------------------------------------------------------------------------

typedef __attribute__((ext_vector_type(2))) float v2f;
typedef __attribute__((ext_vector_type(4))) float v4f;
typedef __attribute__((ext_vector_type(8))) float v8f;

#define DCOLS 4096
#define CHUNK 64            // K columns per staged chunk
#define PITCH 68            // x-tile row pitch (dwords): 16B aligned, 0-conflict
#define PSTRIDE 68          // param region stride: v@0, w@68, zeros@136
#define WAVES 8             // 256 threads / WG
#define ROWS_PER_WG 128

__global__ __launch_bounds__(256) void quad_wmma_main(
    const float* __restrict__ inp,
    const float* __restrict__ vq,    // quad_part (D)
    const float* __restrict__ wl,    // linear_w  (D)
    const float* __restrict__ bias,  // linear_b  (1)
    float* __restrict__ out,
    float* __restrict__ ws,
    int nRows, int ksplit, int segLen)
{
    __shared__ float ldsX[WAVES][2][16 * PITCH];    // double-buffered x tiles
    __shared__ float ldsP[WAVES][2][3 * PSTRIDE];   // [v | w | zeros] x2

    const int lane = threadIdx.x & 31;
    const int wave = threadIdx.x >> 5;
    const int hi   = lane >> 4;          // half-wave: K offset +0 / +2
    const int sel  = lane & 15;
    const int seg  = blockIdx.y;
    const int kBase = seg * segLen;
    const int rowBase = (blockIdx.x * WAVES + wave) * 16;

    // Zero-fill the "A rows 2..15" source region once, both buffers.
    {
        v2f z = {0.0f, 0.0f};
        *(v2f*)&ldsP[wave][0][2 * PSTRIDE + lane * 2] = z;
        *(v2f*)&ldsP[wave][1][2 * PSTRIDE + lane * 2] = z;
    }

    // --- per-lane global staging addresses (fixed bases, immediate deltas) ---
    const int r0 = lane >> 4;            // tile row of this lane's first b128
    const int c0 = (lane * 4) & 63;      // tile col of this lane's b128
    const float* gx = inp + (size_t)(rowBase + r0) * DCOLS + kBase + c0;
    const float* gp = (hi ? wl : vq) + kBase + sel * 4;

    // --- LDS addresses: one scalar pointer per role per buffer, each a
    //     direct GEP on the __shared__ array (addrspace(3) preserved). ---
    float* sX0 = &ldsX[wave][0][r0 * PITCH + c0];
    float* sX1 = &ldsX[wave][1][r0 * PITCH + c0];
    float* sP0 = &ldsP[wave][0][hi * PSTRIDE + sel * 4];
    float* sP1 = &ldsP[wave][1][hi * PSTRIDE + sel * 4];

    const int aoff = (sel == 0 ? 0 : (sel == 1 ? PSTRIDE : 2 * PSTRIDE)) + hi * 2;
    const float* lA0 = &ldsP[wave][0][aoff];
    const float* lA1 = &ldsP[wave][1][aoff];
    const float* lB0 = &ldsX[wave][0][sel * PITCH + hi * 2];
    const float* lB1 = &ldsX[wave][1][sel * PITCH + hi * 2];

    v8f c = {};
    v4f xr[8];
    v4f pr;

#define LOAD_CH(chi) do {                                                   \
        const float* gx_ = gx + (size_t)(chi) * CHUNK;                      \
        pr = *(const v4f*)(gp + (size_t)(chi) * CHUNK);                     \
        _Pragma("unroll")                                                   \
        for (int i = 0; i < 8; ++i)                                         \
            xr[i] = *(const v4f*)(gx_ + (size_t)i * 2 * DCOLS);             \
    } while (0)

#define COMMIT(sx, sp) do {                                                 \
        *(v4f*)(sp) = pr;                                                   \
        _Pragma("unroll")                                                   \
        for (int i = 0; i < 8; ++i)                                         \
            *(v4f*)((sx) + i * 2 * PITCH) = xr[i];                          \
    } while (0)

#define WMMA16(la, lb) do {                                                 \
        _Pragma("unroll")                                                   \
        for (int i = 0; i < 16; ++i) {                                      \
            v2f a = *(const v2f*)((la) + i * 4);                            \
            v2f b = *(const v2f*)((lb) + i * 4);                            \
            c = __builtin_amdgcn_wmma_f32_16x16x4_f32(                      \
                    false, a, false, b, (short)0, c, false, false);         \
        }                                                                   \
    } while (0)

    // ---- software-pipelined main loop; nChunks is even (16 or 64) ----
    LOAD_CH(0);
    const int nChunks = segLen / CHUNK;
    for (int ch = 0; ch < nChunks; ch += 2) {
        COMMIT(sX0, sP0);                      // buffer 0
        LOAD_CH(ch + 1);                       // prefetch next chunk
        WMMA16(lA0, lB0);

        COMMIT(sX1, sP1);                      // buffer 1
        if (ch + 2 < nChunks) LOAD_CH(ch + 2); // prefetch (uniform branch)
        WMMA16(lA1, lB1);
    }

#undef LOAD_CH
#undef COMMIT
#undef WMMA16

    // D[0][n] (qx) in acc elem 0, D[1][n] (lin) in acc elem 1, lanes 0-15
    if (lane < 16) {
        const int row = rowBase + lane;
        const float q = c[0];
        const float l = c[1];
        if (ksplit == 1) {
            out[row] = -(q * q) + l + bias[0];
        } else {
            ws[(size_t)seg * 2 * nRows + row]         = q;
            ws[(size_t)seg * 2 * nRows + nRows + row] = l;
        }
    }
}

__global__ __launch_bounds__(256) void quad_wmma_finalize(
    const float* __restrict__ ws,
    const float* __restrict__ bias,
    float* __restrict__ out,
    int nRows, int ksplit)
{
    int r = blockIdx.x * blockDim.x + threadIdx.x;
    if (r >= nRows) return;
    float q = 0.0f, l = 0.0f;
    for (int s = 0; s < ksplit; ++s) {
        q += ws[(size_t)s * 2 * nRows + r];
        l += ws[(size_t)s * 2 * nRows + nRows + r];
    }
    out[r] = -(q * q) + l + bias[0];
}

extern "C" void kernel_launch(void* const* d_in, const int* in_sizes, int n_in,
                              void* d_out, int out_size, void* d_ws, size_t ws_size,
                              hipStream_t stream) {
    (void)n_in; (void)out_size;
    const float* inp  = (const float*)d_in[0];   // (B, D) fp32
    const float* vq   = (const float*)d_in[1];   // (D, 1) fp32
    const float* wl   = (const float*)d_in[2];   // (1, D) fp32
    const float* bias = (const float*)d_in[3];   // (1,)   fp32
    float* out = (float*)d_out;
    float* ws  = (float*)d_ws;

    const int nRows = in_sizes[0] / DCOLS;       // 16384

    int ksplit = 4;                              // 4096 waves total
    size_t need = (size_t)ksplit * 2 * nRows * sizeof(float);
    if (ws_size < need) ksplit = 1;              // fallback: direct write
    const int segLen = DCOLS / ksplit;

    dim3 grid(nRows / ROWS_PER_WG, ksplit);
    quad_wmma_main<<<grid, 256, 0, stream>>>(inp, vq, wl, bias, out, ws,
                                             nRows, ksplit, segLen);
    if (ksplit > 1) {
        quad_wmma_finalize<<<(nRows + 255) / 256, 256, 0, stream>>>(
            ws, bias, out, nRows, ksplit);
    }
}